// PSA3D_9964324126839
// MI455X (gfx1250) — compile-verified
//
#include <hip/hip_runtime.h>

// ---------------------------------------------------------------------------
// Fused permuted-window 3D self-attention for MI455X (gfx1250, wave32).
// One workgroup per window; all intermediates staged in LDS; all matmuls via
// V_WMMA_F32_16X16X4_F32 (full fp32 precision; 56 GFLOP vs ~400MB HBM traffic
// => memory-bound, need only ~3.3 TF to saturate 23.3 TB/s).
// Q-proj / out-proj register-block 2 M-tiles per weight strip so each global
// (L2) weight fragment feeds two WMMAs -> halves L2 weight traffic.
// ---------------------------------------------------------------------------

typedef float v2f __attribute__((ext_vector_type(2)));
typedef float v8f __attribute__((ext_vector_type(8)));

#define NWIN   512
#define NTOK   512          // queries per window
#define PTOK   64           // kv tokens per window
#define DIMC   192
#define NH     6
#define HD     32
#define LDP    196          // padded LDS row stride (196 % 64 == 4 -> no bank conflicts)
#define SSTR   68           // padded score-row stride (68 % 64 == 4)
#define TABLE  343
#define QSCALE 0.17677669529663687f   // 32^-0.5

__device__ __forceinline__ v8f wmma4(v2f a, v2f b, v8f c) {
  // D = A(16x4 f32) * B(4x16 f32) + C(16x16 f32)
  return __builtin_amdgcn_wmma_f32_16x16x4_f32(false, a, false, b, (short)0, c,
                                               false, false);
}

__global__ __launch_bounds__(256, 1) void psa3d_fused(
    const float* __restrict__ x,     const float* __restrict__ wq,
    const float* __restrict__ bq,    const float* __restrict__ wkv,
    const float* __restrict__ bkv,   const float* __restrict__ btab,
    const float* __restrict__ wproj, const float* __restrict__ bproj,
    const int*   __restrict__ rpi,   float* __restrict__ out)
{
  // r0 aliases: x staging tile (64*LDP) during projections,
  //             3 per-head score tiles (3*64*SSTR) during attention.
  __shared__ float r0[3 * 64 * SSTR];      // 52.2 KB
  __shared__ float ks[64 * LDP];           // 50.2 KB  K (folded, 64 x 192)
  __shared__ float vs[64 * LDP];           // 50.2 KB  V
  __shared__ float qs[64 * LDP];           // 50.2 KB  Q tile (scaled)
  __shared__ float os[64 * LDP];           // 50.2 KB  head-concat AV output
  __shared__ int   rp[64 * 64];            // 16   KB  rpi tile
  __shared__ float bt[TABLE * NH];         // 8.2  KB  bias table

  const int tid  = threadIdx.x;
  const int w    = tid >> 5;       // wave id 0..7
  const int lane = tid & 31;
  const int l16  = lane & 15;
  const int lh   = lane >> 4;      // lane-half: 0 or 1
  const int bw   = blockIdx.x;
  float* xs = r0;

  // stage relative-position bias table once (shared by all heads/chunks)
  for (int e = tid; e < TABLE * NH; e += 256) bt[e] = btab[e];

  // =========================================================================
  // Phase 1: KV projection + 2x2x2 spatial fold into K,V (LDS-resident)
  // =========================================================================
  for (int cc = 0; cc < 8; ++cc) {
    const float* xg = x + ((size_t)bw * NTOK + cc * 64) * DIMC;
    for (int t = 0; t < 12; ++t) {                       // 64x192 tile -> LDS
      int e4 = tid + t * 256;
      int row = e4 / 48, col = (e4 % 48) << 2;
      *(float4*)(xs + row * LDP + col) = *(const float4*)(xg + row * DIMC + col);
    }
    if (cc < 7) __builtin_prefetch(xg + 64 * DIMC + tid * 48, 0, 1);
    __syncthreads();

    for (int tt = w; tt < 12; tt += 8) {                 // 4 Mt x 3 Nt tiles
      int mt = tt / 3, nt = tt % 3;
      v8f acc = {};
      for (int k0 = 0; k0 < DIMC; k0 += 4) {
        v2f a;  { const float* p = xs + (mt*16 + l16)*LDP + k0 + 2*lh;
                  a.x = p[0]; a.y = p[1]; }
        v2f bb; { const float* p = wkv + (k0 + 2*lh)*48 + nt*16 + l16;
                  bb.x = p[0]; bb.y = p[48]; }
        acc = wmma4(a, bb, acc);
      }
      const int c  = nt * 16 + l16;
      const float bv = bkv[c];
      for (int i = 0; i < 8; ++i) {                      // fold-scatter
        int m  = mt * 16 + i + 8 * lh;
        int n  = cc * 64 + m;                            // token = (dd,hh,ww)
        int dd = n >> 6, hh = (n >> 3) & 7, ww = n & 7;
        int p  = ((dd >> 1) << 4) | ((hh >> 1) << 2) | (ww >> 1);
        int ch = (((hh & 1) << 1) | (ww & 1)) * 48 + c;  // (i2b,i2c,c) -> 0..191
        float* dst = (dd & 1) ? vs : ks;                 // i2a: 0 -> K, 1 -> V
        dst[p * LDP + ch] = acc[i] + bv;
      }
    }
    __syncthreads();
  }

  // =========================================================================
  // Phase 2: per 64-query tile: Q proj -> attention -> out projection
  // =========================================================================
  for (int cc = 0; cc < 8; ++cc) {
    const float* xg = x + ((size_t)bw * NTOK + cc * 64) * DIMC;
    for (int t = 0; t < 12; ++t) {
      int e4 = tid + t * 256;
      int row = e4 / 48, col = (e4 % 48) << 2;
      *(float4*)(xs + row * LDP + col) = *(const float4*)(xg + row * DIMC + col);
    }
    const int* rg = rpi + cc * 64 * 64;
    for (int t = 0; t < 4; ++t) {
      int e4 = (tid + t * 256) << 2;
      *(int4*)(rp + e4) = *(const int4*)(rg + e4);
    }
    if (cc < 7) __builtin_prefetch(xg + 64 * DIMC + tid * 48, 0, 1);
    __syncthreads();

    // ---- Q projection: qs = (xs @ wq + bq) * SCALE ----
    // 24 tasks of (2 Mt x 1 Nt); each weight fragment feeds 2 WMMAs.
    for (int tt = w; tt < 24; tt += 8) {
      int nt = tt % 12, mh = tt / 12;                    // mh selects Mt pair
      int m0 = (mh * 2) * 16, m1 = m0 + 16;
      v8f acc0 = {}, acc1 = {};
      for (int k0 = 0; k0 < DIMC; k0 += 4) {
        v2f bb; { const float* p = wq + (k0 + 2*lh)*DIMC + nt*16 + l16;
                  bb.x = p[0]; bb.y = p[DIMC]; }
        v2f a0; { const float* p = xs + (m0 + l16)*LDP + k0 + 2*lh;
                  a0.x = p[0]; a0.y = p[1]; }
        v2f a1; { const float* p = xs + (m1 + l16)*LDP + k0 + 2*lh;
                  a1.x = p[0]; a1.y = p[1]; }
        acc0 = wmma4(a0, bb, acc0);
        acc1 = wmma4(a1, bb, acc1);
      }
      const int c = nt * 16 + l16;
      const float bv = bq[c];
      for (int i = 0; i < 8; ++i) {
        qs[(m0 + i + 8*lh)*LDP + c] = (acc0[i] + bv) * QSCALE;
        qs[(m1 + i + 8*lh)*LDP + c] = (acc1[i] + bv) * QSCALE;
      }
    }
    __syncthreads();

    // ---- attention: 2 groups x (3 heads x 2 waves); waves 6,7 idle ----
    for (int hg = 0; hg < 2; ++hg) {
      const bool act = (w < 6);
      const int sub = w & 1;
      if (act) {                                         // scores = q @ k^T
        const int h = hg * 3 + (w >> 1);
        const int hb = h * HD;
        float* ssh = r0 + (w >> 1) * (64 * SSTR);
        for (int tt = sub; tt < 16; tt += 2) {           // 4 Mt x 4 Nt
          int mt = tt >> 2, nt = tt & 3;
          v8f acc = {};
          for (int k0 = 0; k0 < HD; k0 += 4) {
            v2f a;  { const float* p = qs + (mt*16 + l16)*LDP + hb + k0 + 2*lh;
                      a.x = p[0]; a.y = p[1]; }
            v2f bb; { const float* p = ks + (nt*16 + l16)*LDP + hb + k0 + 2*lh;
                      bb.x = p[0]; bb.y = p[1]; }       // B[k][p] = K[p][k]
            acc = wmma4(a, bb, acc);
          }
          for (int i = 0; i < 8; ++i)
            ssh[(mt*16 + i + 8*lh)*SSTR + nt*16 + l16] = acc[i];
        }
      }
      __syncthreads();
      if (act) {                                         // bias + softmax
        const int h = hg * 3 + (w >> 1);
        float* ssh = r0 + (w >> 1) * (64 * SSTR);
        for (int r = sub * 32; r < sub * 32 + 32; ++r) {
          float s0 = ssh[r*SSTR + lane]      + bt[rp[r*64 + lane]      * NH + h];
          float s1 = ssh[r*SSTR + lane + 32] + bt[rp[r*64 + lane + 32] * NH + h];
          float mx = fmaxf(s0, s1);
          for (int o = 16; o; o >>= 1) mx = fmaxf(mx, __shfl_xor(mx, o, 32));
          float e0 = __expf(s0 - mx), e1 = __expf(s1 - mx);
          float sm = e0 + e1;
          for (int o = 16; o; o >>= 1) sm += __shfl_xor(sm, o, 32);
          float inv = 1.0f / sm;
          ssh[r*SSTR + lane]      = e0 * inv;
          ssh[r*SSTR + lane + 32] = e1 * inv;
        }
      }
      __syncthreads();
      if (act) {                                         // os[:,h] = attn @ v
        const int h = hg * 3 + (w >> 1);
        const int hb = h * HD;
        float* ssh = r0 + (w >> 1) * (64 * SSTR);
        for (int tt = sub; tt < 8; tt += 2) {            // 4 Mt x 2 Nt
          int mt = tt >> 1, nt = tt & 1;
          v8f acc = {};
          for (int k0 = 0; k0 < PTOK; k0 += 4) {
            v2f a;  { const float* p = ssh + (mt*16 + l16)*SSTR + k0 + 2*lh;
                      a.x = p[0]; a.y = p[1]; }
            v2f bb; { const float* p = vs + (k0 + 2*lh)*LDP + hb + nt*16 + l16;
                      bb.x = p[0]; bb.y = p[LDP]; }
            acc = wmma4(a, bb, acc);
          }
          for (int i = 0; i < 8; ++i)
            os[(mt*16 + i + 8*lh)*LDP + hb + nt*16 + l16] = acc[i];
        }
      }
      __syncthreads();
    }

    // ---- output projection: out = os @ wproj + bproj ----
    // Same 2-Mt register blocking as Q-proj.
    float* og = out + ((size_t)bw * NTOK + cc * 64) * DIMC;
    for (int tt = w; tt < 24; tt += 8) {
      int nt = tt % 12, mh = tt / 12;
      int m0 = (mh * 2) * 16, m1 = m0 + 16;
      v8f acc0 = {}, acc1 = {};
      for (int k0 = 0; k0 < DIMC; k0 += 4) {
        v2f bb; { const float* p = wproj + (k0 + 2*lh)*DIMC + nt*16 + l16;
                  bb.x = p[0]; bb.y = p[DIMC]; }
        v2f a0; { const float* p = os + (m0 + l16)*LDP + k0 + 2*lh;
                  a0.x = p[0]; a0.y = p[1]; }
        v2f a1; { const float* p = os + (m1 + l16)*LDP + k0 + 2*lh;
                  a1.x = p[0]; a1.y = p[1]; }
        acc0 = wmma4(a0, bb, acc0);
        acc1 = wmma4(a1, bb, acc1);
      }
      const int c = nt * 16 + l16;
      const float bv = bproj[c];
      for (int i = 0; i < 8; ++i) {
        og[(m0 + i + 8*lh)*DIMC + c] = acc0[i] + bv;
        og[(m1 + i + 8*lh)*DIMC + c] = acc1[i] + bv;
      }
    }
    __syncthreads();
  }
}

extern "C" void kernel_launch(void* const* d_in, const int* in_sizes, int n_in,
                              void* d_out, int out_size, void* d_ws, size_t ws_size,
                              hipStream_t stream) {
  (void)in_sizes; (void)n_in; (void)out_size; (void)d_ws; (void)ws_size;
  const float* x     = (const float*)d_in[0];
  const float* wq    = (const float*)d_in[1];
  const float* bq    = (const float*)d_in[2];
  const float* wkv   = (const float*)d_in[3];
  const float* bkv   = (const float*)d_in[4];
  const float* btab  = (const float*)d_in[5];
  const float* wproj = (const float*)d_in[6];
  const float* bproj = (const float*)d_in[7];
  const int*   rpi   = (const int*)d_in[8];
  float* out = (float*)d_out;
  psa3d_fused<<<NWIN, 256, 0, stream>>>(x, wq, bq, wkv, bkv, btab, wproj, bproj,
                                        rpi, out);
}